// FFTConvReservoir_15238543966956
// MI455X (gfx1250) — compile-verified
//
#include <hip/hip_runtime.h>
#include <math.h>

// ---------------------------------------------------------------------------
// FFT circular convolution on MI455X (gfx1250), four-step WMMA FFT.
//   y = tanh( ifft(fft(u) * fft(K)).real + D*u )
// B=8, H=256, L=8192.  L = 64 x 128 four-step decomposition:
//   n = 128*n1 + n2 ; k = k1 + 64*k2
//   fwd : S[k1][k2] = sum_n2 w8192^(n2 k1) wN2^(n2 k2) sum_n1 x[n] wN1^(n1 k1)
//   inv : x[n] = (1/L) sum_k1 wN1^(-n1 k1) w8192^(-n2 k1) sum_k2 S wN2^(-n2 k2)
// Spectra stay in permuted [k1][k2] layout (pointwise mult is order-agnostic).
// Complex-pair trick: conv is linear in u, so u[2b] + i*u[2b+1] shares one
// transform; y(2b)=re, y(2b+1)=im.  All matmuls run on V_WMMA_F32_16X16X4_F32.
// Each wave register-blocks a 32x32 output block (2x2 WMMA tiles) so operand
// loads are amortized: 10 loads per 16 WMMAs.  DFT operands are AS1-qualified
// (pure global_load, LOADcnt only); LDS planes are AS3-qualified (pure ds_*,
// DScnt only) so the two dependency domains stay independent (no flat_*).
// ---------------------------------------------------------------------------

typedef __attribute__((ext_vector_type(2))) float v2f;
typedef __attribute__((ext_vector_type(8))) float v8f;

typedef __attribute__((address_space(1))) float ga_f32;   // global
typedef __attribute__((address_space(3))) float ls_f32;   // LDS

#define B_   8
#define H_   256
#define L_   8192
#define N1_  64
#define N2_  128
#define LDJ  132          // padded LDS row stride (floats) to spread banks
#define TWO_PI 6.28318530717958647692f

// ---- workspace layout (float offsets); total ~17.3 MB ---------------------
#define OFF_F64R   0                        // fwd DFT64: cos
#define OFF_F64I   (OFF_F64R + 64*64)       //            -sin
#define OFF_F64N   (OFF_F64I + 64*64)       //            +sin (negated imag)
#define OFF_G64R   (OFF_F64N + 64*64)       // inv DFT64 (scaled by 1/L)
#define OFF_G64I   (OFF_G64R + 64*64)
#define OFF_G64N   (OFF_G64I + 64*64)
#define OFF_F128R  (OFF_G64N + 64*64)       // fwd DFT128
#define OFF_F128I  (OFF_F128R + 128*128)
#define OFF_F128N  (OFF_F128I + 128*128)
#define OFF_G128R  (OFF_F128N + 128*128)    // inv DFT128 (unscaled)
#define OFF_G128I  (OFF_G128R + 128*128)
#define OFF_G128N  (OFF_G128I + 128*128)
#define OFF_KHAT   (OFF_G128N + 128*128)    // H * (re[8192], im[8192])

// ---- gfx1250 async global->LDS path (guarded; falls back to plain loads) --
#if defined(__gfx1250__) && __has_builtin(__builtin_amdgcn_global_load_async_to_lds_b32)
#define HAVE_ASYNC_LDS 1
typedef __attribute__((address_space(1))) int as1_int;
typedef __attribute__((address_space(3))) int as3_int;
__device__ __forceinline__ void async_ld32(const float* g, float* l) {
  __builtin_amdgcn_global_load_async_to_lds_b32((as1_int*)g, (as3_int*)l, 0, 0);
}
__device__ __forceinline__ void wait_async0() {
#if __has_builtin(__builtin_amdgcn_s_wait_asynccnt)
  __builtin_amdgcn_s_wait_asynccnt(0);
#else
  asm volatile("s_wait_asynccnt 0" ::: "memory");
#endif
}
#else
#define HAVE_ASYNC_LDS 0
#endif

// ---------------------------------------------------------------------------
// WMMA f32 16x16x4 helpers.  Operand layouts per CDNA5 ISA 7.12.2:
//  A (16x4): lane&15 = M ; VGPR0 holds K = 2*(lane>>4), VGPR1 holds K+1
//  B (4x16): lane&15 = N ; VGPR0 holds K = 2*(lane>>4), VGPR1 holds K+1
//  C/D     : lane&15 = N ; VGPR i holds M = i + 8*(lane>>4)
// ---------------------------------------------------------------------------
__device__ __forceinline__ v8f wmma4(v2f a, v2f b, v8f c) {
  return __builtin_amdgcn_wmma_f32_16x16x4_f32(false, a, false, b,
                                               (short)0, c, false, false);
}

template <typename P>
__device__ __forceinline__ v2f load_a_t(P p0, int ld, int m0, int k0,
                                        int lane) {
  int m = m0 + (lane & 15);
  int k = k0 + ((lane >> 4) << 1);
  P p = p0 + m * ld + k;
  v2f r; r.x = p[0]; r.y = p[1];
  return r;
}

template <typename P>
__device__ __forceinline__ v2f load_b_t(P p0, int ld, int k0, int n0,
                                        int lane) {
  int n = n0 + (lane & 15);
  int k = k0 + ((lane >> 4) << 1);
  P p = p0 + k * ld + n;
  v2f r; r.x = p[0]; r.y = p[ld];
  return r;
}

// Store a complex 16x16 C/D tile to LDS, optionally applying the four-step
// twiddle  exp(TW * i * 2*pi * (m*n)/L)  elementwise (TW in {-1,0,+1}).
template <int TW>
__device__ __forceinline__ void store_cd(ls_f32* ore, ls_f32* oim, int ld,
                                         int m0, int n0, int lane,
                                         v8f cre, v8f cim) {
  int n  = n0 + (lane & 15);
  int mb = m0 + ((lane >> 4) << 3);
#pragma unroll
  for (int i = 0; i < 8; ++i) {
    float re = cre[i], im = cim[i];
    if (TW != 0) {
      int r = ((mb + i) * n) & (L_ - 1);                // (m*n) mod 8192
      float th = (float)TW * (TWO_PI / (float)L_) * (float)r;
      float s, c;
      __sincosf(th, &s, &c);
      float re2 = re * c - im * s;
      im = im * c + re * s;
      re = re2;
    }
    ore[(mb + i) * ld + n] = re;
    oim[(mb + i) * ld + n] = im;
  }
}

// out[64][128] = F(64x64, A-operand, global) x data(64x128, B-operand, LDS)
// complex matmul via 4 real WMMAs per K-step per tile; fn = -imag(F).
// Each wave owns a 32x32 block: m0 in {0,32}, n0 in {0,32,64,96}.
template <int TW>
__device__ void col_fft(const float* fr_, const float* fi_, const float* fn_,
                        const float* dre_, const float* dim__,
                        float* ore_, float* oim_, int lane, int wave) {
  const ga_f32* fr = (const ga_f32*)fr_;
  const ga_f32* fi = (const ga_f32*)fi_;
  const ga_f32* fn = (const ga_f32*)fn_;
  const ls_f32* dre = (const ls_f32*)dre_;
  const ls_f32* dim_ = (const ls_f32*)dim__;
  ls_f32* ore = (ls_f32*)ore_;
  ls_f32* oim = (ls_f32*)oim_;
  int m0 = (wave >> 2) << 5;
  int n0 = (wave & 3) << 5;
  v8f ar00 = {}, ai00 = {}, ar01 = {}, ai01 = {};
  v8f ar10 = {}, ai10 = {}, ar11 = {}, ai11 = {};
#pragma unroll 2
  for (int k0 = 0; k0 < N1_; k0 += 4) {
    v2f far0 = load_a_t(fr, N1_, m0,      k0, lane);
    v2f fai0 = load_a_t(fi, N1_, m0,      k0, lane);
    v2f fan0 = load_a_t(fn, N1_, m0,      k0, lane);
    v2f far1 = load_a_t(fr, N1_, m0 + 16, k0, lane);
    v2f fai1 = load_a_t(fi, N1_, m0 + 16, k0, lane);
    v2f fan1 = load_a_t(fn, N1_, m0 + 16, k0, lane);
    v2f dbr0 = load_b_t(dre,  LDJ, k0, n0,      lane);
    v2f dbi0 = load_b_t(dim_, LDJ, k0, n0,      lane);
    v2f dbr1 = load_b_t(dre,  LDJ, k0, n0 + 16, lane);
    v2f dbi1 = load_b_t(dim_, LDJ, k0, n0 + 16, lane);
    ar00 = wmma4(far0, dbr0, ar00);  ar00 = wmma4(fan0, dbi0, ar00);
    ai00 = wmma4(far0, dbi0, ai00);  ai00 = wmma4(fai0, dbr0, ai00);
    ar01 = wmma4(far0, dbr1, ar01);  ar01 = wmma4(fan0, dbi1, ar01);
    ai01 = wmma4(far0, dbi1, ai01);  ai01 = wmma4(fai0, dbr1, ai01);
    ar10 = wmma4(far1, dbr0, ar10);  ar10 = wmma4(fan1, dbi0, ar10);
    ai10 = wmma4(far1, dbi0, ai10);  ai10 = wmma4(fai1, dbr0, ai10);
    ar11 = wmma4(far1, dbr1, ar11);  ar11 = wmma4(fan1, dbi1, ar11);
    ai11 = wmma4(far1, dbi1, ai11);  ai11 = wmma4(fai1, dbr1, ai11);
  }
  store_cd<TW>(ore, oim, LDJ, m0,      n0,      lane, ar00, ai00);
  store_cd<TW>(ore, oim, LDJ, m0,      n0 + 16, lane, ar01, ai01);
  store_cd<TW>(ore, oim, LDJ, m0 + 16, n0,      lane, ar10, ai10);
  store_cd<TW>(ore, oim, LDJ, m0 + 16, n0 + 16, lane, ar11, ai11);
}

// out[64][128] = data(64x128, A-operand, LDS) x G(128x128, B-operand, global)
template <int TW>
__device__ void row_fft(const float* gr_, const float* gi_, const float* gn_,
                        const float* dre_, const float* dim__,
                        float* ore_, float* oim_, int lane, int wave) {
  const ga_f32* gr = (const ga_f32*)gr_;
  const ga_f32* gi = (const ga_f32*)gi_;
  const ga_f32* gn = (const ga_f32*)gn_;
  const ls_f32* dre = (const ls_f32*)dre_;
  const ls_f32* dim_ = (const ls_f32*)dim__;
  ls_f32* ore = (ls_f32*)ore_;
  ls_f32* oim = (ls_f32*)oim_;
  int m0 = (wave >> 2) << 5;
  int n0 = (wave & 3) << 5;
  v8f ar00 = {}, ai00 = {}, ar01 = {}, ai01 = {};
  v8f ar10 = {}, ai10 = {}, ar11 = {}, ai11 = {};
#pragma unroll 2
  for (int k0 = 0; k0 < N2_; k0 += 4) {
    v2f dar0 = load_a_t(dre,  LDJ, m0,      k0, lane);
    v2f dai0 = load_a_t(dim_, LDJ, m0,      k0, lane);
    v2f dar1 = load_a_t(dre,  LDJ, m0 + 16, k0, lane);
    v2f dai1 = load_a_t(dim_, LDJ, m0 + 16, k0, lane);
    v2f gbr0 = load_b_t(gr, N2_, k0, n0,      lane);
    v2f gbi0 = load_b_t(gi, N2_, k0, n0,      lane);
    v2f gbn0 = load_b_t(gn, N2_, k0, n0,      lane);
    v2f gbr1 = load_b_t(gr, N2_, k0, n0 + 16, lane);
    v2f gbi1 = load_b_t(gi, N2_, k0, n0 + 16, lane);
    v2f gbn1 = load_b_t(gn, N2_, k0, n0 + 16, lane);
    ar00 = wmma4(dar0, gbr0, ar00);  ar00 = wmma4(dai0, gbn0, ar00);
    ai00 = wmma4(dar0, gbi0, ai00);  ai00 = wmma4(dai0, gbr0, ai00);
    ar01 = wmma4(dar0, gbr1, ar01);  ar01 = wmma4(dai0, gbn1, ar01);
    ai01 = wmma4(dar0, gbi1, ai01);  ai01 = wmma4(dai0, gbr1, ai01);
    ar10 = wmma4(dar1, gbr0, ar10);  ar10 = wmma4(dai1, gbn0, ar10);
    ai10 = wmma4(dar1, gbi0, ai10);  ai10 = wmma4(dai1, gbr0, ai10);
    ar11 = wmma4(dar1, gbr1, ar11);  ar11 = wmma4(dai1, gbn1, ar11);
    ai11 = wmma4(dar1, gbi1, ai11);  ai11 = wmma4(dai1, gbr1, ai11);
  }
  store_cd<TW>(ore, oim, LDJ, m0,      n0,      lane, ar00, ai00);
  store_cd<TW>(ore, oim, LDJ, m0,      n0 + 16, lane, ar01, ai01);
  store_cd<TW>(ore, oim, LDJ, m0 + 16, n0,      lane, ar10, ai10);
  store_cd<TW>(ore, oim, LDJ, m0 + 16, n0 + 16, lane, ar11, ai11);
}

// ---------------------------------------------------------------------------
// Kernel 1: build DFT matrices (runs once per launch; trivial cost).
// ---------------------------------------------------------------------------
__global__ void __launch_bounds__(256) build_dft_kernel(float* ws) {
  int t = blockIdx.x * blockDim.x + threadIdx.x;
  if (t < 64 * 64) {
    int m = t >> 6, n = t & 63;
    int r = (m * n) & 63;
    float th = TWO_PI * (float)r / 64.0f;
    float s, c;
    sincosf(th, &s, &c);
    ws[OFF_F64R + t] = c;      // w64^(mn) = cos - i sin
    ws[OFF_F64I + t] = -s;
    ws[OFF_F64N + t] = s;
    float sc = 1.0f / (float)L_;            // fold 1/L into the LAST stage
    ws[OFF_G64R + t] = c * sc;              // w64^(-mn)/L = (cos + i sin)/L
    ws[OFF_G64I + t] = s * sc;
    ws[OFF_G64N + t] = -s * sc;
  }
  if (t < 128 * 128) {
    int m = t >> 7, n = t & 127;
    int r = (m * n) & 127;
    float th = TWO_PI * (float)r / 128.0f;
    float s, c;
    sincosf(th, &s, &c);
    ws[OFF_F128R + t] = c;
    ws[OFF_F128I + t] = -s;
    ws[OFF_F128N + t] = s;
    ws[OFF_G128R + t] = c;
    ws[OFF_G128I + t] = s;
    ws[OFF_G128N + t] = -s;
  }
}

// ---------------------------------------------------------------------------
// Kernel 2: Khat[h] = FFT(K[h]) in permuted [k1][k2] layout. One WG per h.
// ---------------------------------------------------------------------------
__global__ void __launch_bounds__(256) khat_kernel(const float* __restrict__ K,
                                                   float* __restrict__ ws) {
  extern __shared__ float smem[];
  float* Are = smem;
  float* Aim = Are + 64 * LDJ;
  float* Bre = Aim + 64 * LDJ;
  float* Bim = Bre + 64 * LDJ;
  int tid = threadIdx.x, lane = tid & 31, wave = tid >> 5;
  int h = blockIdx.x;

  for (int idx = tid; idx < L_; idx += 256) {
    int n1 = idx >> 7, n2 = idx & 127;
#if HAVE_ASYNC_LDS
    async_ld32(K + (size_t)h * L_ + idx, &Are[n1 * LDJ + n2]);
#else
    Are[n1 * LDJ + n2] = K[(size_t)h * L_ + idx];
#endif
    Aim[n1 * LDJ + n2] = 0.0f;
  }
#if HAVE_ASYNC_LDS
  wait_async0();
#endif
  __syncthreads();
  col_fft<-1>(ws + OFF_F64R, ws + OFF_F64I, ws + OFF_F64N,
              Are, Aim, Bre, Bim, lane, wave);
  __syncthreads();
  row_fft<0>(ws + OFF_F128R, ws + OFF_F128I, ws + OFF_F128N,
             Bre, Bim, Are, Aim, lane, wave);
  __syncthreads();
  float* kh = ws + OFF_KHAT + (size_t)h * (2 * L_);
  for (int idx = tid; idx < L_; idx += 256) {
    int n1 = idx >> 7, n2 = idx & 127;
    kh[idx]      = Are[n1 * LDJ + n2];
    kh[L_ + idx] = Aim[n1 * LDJ + n2];
  }
}

// ---------------------------------------------------------------------------
// Kernel 3: main conv. One WG per (h, batch-pair). 1024 WGs, 8 waves each.
// ---------------------------------------------------------------------------
__global__ void __launch_bounds__(256) fftconv_kernel(
    const float* __restrict__ u, const float* __restrict__ D,
    const float* __restrict__ ws, float* __restrict__ out) {
  extern __shared__ float smem[];
  float* Are = smem;
  float* Aim = Are + 64 * LDJ;
  float* Bre = Aim + 64 * LDJ;
  float* Bim = Bre + 64 * LDJ;
  int tid = threadIdx.x, lane = tid & 31, wave = tid >> 5;
  int h  = blockIdx.x >> 2;
  int pr = blockIdx.x & 3;
  int b0 = 2 * pr, b1 = 2 * pr + 1;

  const float* kh  = ws + OFF_KHAT + (size_t)h * (2 * L_);
  const float* u0p = u + ((size_t)b0 * H_ + h) * L_;
  const float* u1p = u + ((size_t)b1 * H_ + h) * L_;

  // warm L2/WGP$ for the Khat row (64KB): one 256B-granular prefetch/thread
  __builtin_prefetch(kh + (size_t)tid * 64, 0, 0);

  // load complex pair u[b0] + i*u[b1] into LDS, natural order n = 128*n1+n2
  for (int idx = tid; idx < L_; idx += 256) {
    int n1 = idx >> 7, n2 = idx & 127;
#if HAVE_ASYNC_LDS
    async_ld32(u0p + idx, &Are[n1 * LDJ + n2]);
    async_ld32(u1p + idx, &Aim[n1 * LDJ + n2]);
#else
    Are[n1 * LDJ + n2] = u0p[idx];
    Aim[n1 * LDJ + n2] = u1p[idx];
#endif
  }
#if HAVE_ASYNC_LDS
  wait_async0();
#endif
  __syncthreads();

  // ---- forward: col DFT64 (+twiddle w^-mn), then row DFT128 ----
  col_fft<-1>(ws + OFF_F64R, ws + OFF_F64I, ws + OFF_F64N,
              Are, Aim, Bre, Bim, lane, wave);
  __syncthreads();
  row_fft<0>(ws + OFF_F128R, ws + OFF_F128I, ws + OFF_F128N,
             Bre, Bim, Are, Aim, lane, wave);
  __syncthreads();

  // ---- pointwise spectrum product with Khat (same permuted layout) ----
  for (int idx = tid; idx < L_; idx += 256) {
    int n1 = idx >> 7, n2 = idx & 127;
    int a = n1 * LDJ + n2;
    float xr = Are[a], xi = Aim[a];
    float kr = kh[idx], ki = kh[L_ + idx];
    Bre[a] = xr * kr - xi * ki;
    Bim[a] = xr * ki + xi * kr;
  }
  __syncthreads();

  // ---- inverse: row IDFT128 (+twiddle w^+mn), then scaled col IDFT64 ----
  row_fft<1>(ws + OFF_G128R, ws + OFF_G128I, ws + OFF_G128N,
             Bre, Bim, Are, Aim, lane, wave);
  __syncthreads();
  col_fft<0>(ws + OFF_G64R, ws + OFF_G64I, ws + OFF_G64N,
             Are, Aim, Bre, Bim, lane, wave);
  __syncthreads();

  // ---- epilogue: unpack pair, skip connection, tanh ----
  float Dh = D[h];
  float* o0 = out + ((size_t)b0 * H_ + h) * L_;
  float* o1 = out + ((size_t)b1 * H_ + h) * L_;
  for (int idx = tid; idx < L_; idx += 256) {
    int n1 = idx >> 7, n2 = idx & 127;
    int a = n1 * LDJ + n2;
    o0[idx] = tanhf(Bre[a] + Dh * u0p[idx]);
    o1[idx] = tanhf(Bim[a] + Dh * u1p[idx]);
  }
}

// ---------------------------------------------------------------------------
extern "C" void kernel_launch(void* const* d_in, const int* in_sizes, int n_in,
                              void* d_out, int out_size, void* d_ws,
                              size_t ws_size, hipStream_t stream) {
  const float* u = (const float*)d_in[0];   // (B,H,L) f32
  const float* K = (const float*)d_in[1];   // (H,L)   f32
  const float* D = (const float*)d_in[2];   // (H,)    f32
  float* out = (float*)d_out;               // (B,H,L) f32
  float* ws  = (float*)d_ws;                // needs ~17.3 MB

  size_t shmem = (size_t)4 * 64 * LDJ * sizeof(float);  // 132 KB -> 2 WG/WGP

  hipLaunchKernelGGL(build_dft_kernel, dim3(64), dim3(256), 0, stream, ws);
  hipLaunchKernelGGL(khat_kernel, dim3(H_), dim3(256), shmem, stream, K, ws);
  hipLaunchKernelGGL(fftconv_kernel, dim3(H_ * 4), dim3(256), shmem, stream,
                     u, D, ws, out);
}